// BlockAttnRes_47957604827781
// MI455X (gfx1250) — compile-verified
//
#include <hip/hip_runtime.h>

// Shapes fixed by the reference setup_inputs():
//   values       [16, 4, 2048, 1024] f32
//   w_query      [24, 1024] f32
//   key_pos_bias [24, 1024] f32
//   position     scalar int (=16)
// Outputs: routed [4,2048,1024] f32  then  alpha [4,2048,16] f32 (concat flat).

#define NB      16              // history slots (== position)
#define BB      4
#define TT      2048
#define HH      1024
#define NTOK    (BB * TT)       // 8192 tokens
#define TPB_TOK 8               // tokens per block
#define GRID    (NTOK / TPB_TOK)
#define THREADS 256             // 8 waves of 32
#define SCALE_INV 0.03125f      // 1 / (sqrt(1024) * tau)

// LDS layout (floats): buf[2][16][1024] | ss[16] | qv[16] | qb[16]
#define SMEM_FLOATS (2 * NB * HH + 48)
#define SMEM_BYTES  (SMEM_FLOATS * 4)

__device__ __forceinline__ float dot4(const float4 a, const float4 b) {
  return a.x * b.x + a.y * b.y + a.z * b.z + a.w * b.w;
}

// CDNA5 async copy global -> LDS. Per ISA 15.18.3 (opcode 98):
//   each lane copies global vaddr+off to LDS vdst+off  (16 bytes/lane, ASYNCcnt)
// IOFFSET is added on BOTH sides, so one base pair covers a 4KB row in 8 ops.
#define ASYNC_B128(LDS, GPTR, OFFS)                                            \
  asm volatile("global_load_async_to_lds_b128 %0, %1, off offset:" OFFS        \
               " th:TH_LOAD_NT"                                                \
               :: "v"(LDS), "v"(GPTR) : "memory")

__device__ __forceinline__ void async_row_4k(unsigned lds, const float* g) {
  ASYNC_B128(lds, g, "0");
  ASYNC_B128(lds, g, "512");
  ASYNC_B128(lds, g, "1024");
  ASYNC_B128(lds, g, "1536");
  ASYNC_B128(lds, g, "2048");
  ASYNC_B128(lds, g, "2560");
  ASYNC_B128(lds, g, "3072");
  ASYNC_B128(lds, g, "3584");
}

__global__ void __launch_bounds__(THREADS)
block_attn_res_kernel(const float* __restrict__ values,
                      const float* __restrict__ w_query,
                      const float* __restrict__ key_pos_bias,
                      const int*   __restrict__ position,
                      float* __restrict__ routed,
                      float* __restrict__ alpha_out) {
  extern __shared__ __align__(16) float smem[];
  float* buf0 = smem;
  float* buf1 = smem + NB * HH;
  float* ss_l = smem + 2 * NB * HH;   // [16]
  float* qv_l = ss_l + 16;            // [16]
  float* qb_l = ss_l + 32;            // [16]

  const int tid  = threadIdx.x;
  const int lane = tid & 31;
  const int wave = tid >> 5;          // 0..7; owns history rows 2w, 2w+1
  const int n0   = 2 * wave;
  const int n1   = n0 + 1;
  const int pos  = *position;         // 16

  // Query slices in registers (phase-A layout: element k*128 + lane*4).
  float4 q[8];
  const float* qg = w_query + (size_t)pos * HH + lane * 4;
#pragma unroll
  for (int k = 0; k < 8; ++k) q[k] = *(const float4*)(qg + k * 128);

  // qb[n] = dot(query, key_pos_bias[n])  — token-independent, once per block.
  {
    float b0 = 0.f, b1 = 0.f;
    const float* g0 = key_pos_bias + (size_t)n0 * HH + lane * 4;
    const float* g1 = key_pos_bias + (size_t)n1 * HH + lane * 4;
#pragma unroll
    for (int k = 0; k < 8; ++k) {
      b0 += dot4(q[k], *(const float4*)(g0 + k * 128));
      b1 += dot4(q[k], *(const float4*)(g1 + k * 128));
    }
#pragma unroll
    for (int m = 16; m >= 1; m >>= 1) {
      b0 += __shfl_xor(b0, m, 32);
      b1 += __shfl_xor(b1, m, 32);
    }
    if (lane == 0) { qb_l[n0] = b0; qb_l[n1] = b1; }
  }
  __syncthreads();

  const int tok_base = blockIdx.x * TPB_TOK;

  // Issue this wave's 2 rows (16 async b128 ops) for one token into bufc.
  auto issue = [&](int token, float* bufc) {
    const unsigned lds0 = (unsigned)(size_t)bufc + (unsigned)(n0 * 4096) + lane * 16;
    const float* g0 = values + ((size_t)n0 * NTOK + token) * HH + lane * 4;
    const float* g1 = values + ((size_t)n1 * NTOK + token) * HH + lane * 4;
    async_row_4k(lds0, g0);
    async_row_4k(lds0 + 4096u, g1);
  };

  issue(tok_base, buf0);  // prologue prefetch

  for (int i = 0; i < TPB_TOK; ++i) {
    float* bufc = (i & 1) ? buf1 : buf0;
    float* bufn = (i & 1) ? buf0 : buf1;

    if (i + 1 < TPB_TOK) {
      issue(tok_base + i + 1, bufn);               // prefetch next token
      asm volatile("s_wait_asynccnt 16" ::: "memory");  // current 16 done, next in flight
    } else {
      asm volatile("s_wait_asynccnt 0" ::: "memory");
    }
    __syncthreads();  // all waves' rows visible in LDS

    // ---- Phase A: per-row ss (sum v^2) and qv (sum q*v), wave-local reduce.
    {
      const float* r0 = bufc + n0 * HH + lane * 4;
      const float* r1 = bufc + n1 * HH + lane * 4;
      float ss0 = 0.f, qv0 = 0.f, ss1 = 0.f, qv1 = 0.f;
#pragma unroll
      for (int k = 0; k < 8; ++k) {
        float4 v0 = *(const float4*)(r0 + k * 128);
        float4 v1 = *(const float4*)(r1 + k * 128);
        ss0 += dot4(v0, v0);  qv0 += dot4(q[k], v0);
        ss1 += dot4(v1, v1);  qv1 += dot4(q[k], v1);
      }
#pragma unroll
      for (int m = 16; m >= 1; m >>= 1) {
        ss0 += __shfl_xor(ss0, m, 32);  qv0 += __shfl_xor(qv0, m, 32);
        ss1 += __shfl_xor(ss1, m, 32);  qv1 += __shfl_xor(qv1, m, 32);
      }
      if (lane == 0) { ss_l[n0] = ss0; qv_l[n0] = qv0; ss_l[n1] = ss1; qv_l[n1] = qv1; }
    }
    __syncthreads();

    // ---- Phase B: softmax over 16 slots (redundant per thread) + weighted sum.
    {
      const int token = tok_base + i;
      float sc[NB];
      float mx = -3.0e38f;
#pragma unroll
      for (int n = 0; n < NB; ++n) {
        float inv = rsqrtf(ss_l[n] * (1.0f / HH) + 1e-6f);
        sc[n] = (inv * qv_l[n] + qb_l[n]) * SCALE_INV;
        mx = fmaxf(mx, sc[n]);
      }
      float sum = 0.f;
#pragma unroll
      for (int n = 0; n < NB; ++n) { sc[n] = __expf(sc[n] - mx); sum += sc[n]; }
      const float rs = 1.0f / sum;

      float4 acc = make_float4(0.f, 0.f, 0.f, 0.f);
      const float* base = bufc + tid * 4;
#pragma unroll
      for (int n = 0; n < NB; ++n) {
        const float a = sc[n] * rs;
        sc[n] = a;                       // keep normalized alpha for output
        float4 v = *(const float4*)(base + n * HH);
        acc.x += a * v.x; acc.y += a * v.y; acc.z += a * v.z; acc.w += a * v.w;
      }
      *(float4*)(routed + (size_t)token * HH + tid * 4) = acc;

      if (tid < NB) {  // alpha [token, n] — select own slot without dynamic indexing
        float a = sc[0];
#pragma unroll
        for (int n = 1; n < NB; ++n) a = (tid == n) ? sc[n] : a;
        alpha_out[(size_t)token * NB + tid] = a;
      }
    }
    __syncthreads();  // bufc may be overwritten by next iteration's prefetch
  }
}

extern "C" void kernel_launch(void* const* d_in, const int* in_sizes, int n_in,
                              void* d_out, int out_size, void* d_ws, size_t ws_size,
                              hipStream_t stream) {
  (void)in_sizes; (void)n_in; (void)out_size; (void)d_ws; (void)ws_size;
  const float* values       = (const float*)d_in[0];
  const float* w_query      = (const float*)d_in[1];
  const float* key_pos_bias = (const float*)d_in[2];
  const int*   position     = (const int*)d_in[3];
  float* routed    = (float*)d_out;
  float* alpha_out = (float*)d_out + (size_t)NTOK * HH;

  block_attn_res_kernel<<<dim3(GRID), dim3(THREADS), SMEM_BYTES, stream>>>(
      values, w_query, key_pos_bias, position, routed, alpha_out);
}